// ConvPartial_1116691497441
// MI455X (gfx1250) — compile-verified
//
#include <hip/hip_runtime.h>
#include <hip/hip_bf16.h>

typedef float v2f __attribute__((ext_vector_type(2)));
typedef float v8f __attribute__((ext_vector_type(8)));
typedef unsigned int u32x4 __attribute__((ext_vector_type(4)));
typedef int i32x4 __attribute__((ext_vector_type(4)));
typedef int i32x8 __attribute__((ext_vector_type(8)));

#define E_TOT 32768
#define TILE  16
#define LN_EPS 1e-5f
#define SLOPE 0.2f

__device__ __forceinline__ float lrelu(float x) { return x >= 0.f ? x : SLOPE * x; }

// Issue one TDM load (descriptor groups per CDNA5 ISA).
__device__ __forceinline__ void tdm_load(u32x4 g0, i32x8 g1, i32x4 g2, i32x4 g3) {
#if __clang_major__ >= 23
    i32x8 g4 = {0, 0, 0, 0, 0, 0, 0, 0};
    __builtin_amdgcn_tensor_load_to_lds(g0, g1, g2, g3, g4, 0);
#else
    __builtin_amdgcn_tensor_load_to_lds(g0, g1, g2, g3, 0);
#endif
}

// One workgroup = 16 edges. 256 threads = 8 waves (wave32).
__global__ __launch_bounds__(256) void equiconv_fused(
    const float* __restrict__ rg,
    const float* __restrict__ B00g,   // [E]
    const float* __restrict__ B01g,   // [E,9]  (basis_00_11, indexed by ro)
    const float* __restrict__ B10g,   // [E,9]  (basis_11_00, indexed by ri)
    const float* __restrict__ B11g,   // [E,9,9,9] (ro,ri,f)
    const float* __restrict__ hg,     // [N,8,10]
    const int*   __restrict__ sidx,   // [E]
    const float* __restrict__ w0g,    // [4,32]
    const float* __restrict__ g0g,    // [4,32]
    const float* __restrict__ be0g,   // [4,32]
    const float* __restrict__ w1g,    // [4,32,32]  (k,o,i)
    const float* __restrict__ b1g,    // [4,32]
    const float* __restrict__ g1g,    // [4,32]
    const float* __restrict__ be1g,   // [4,32]
    const float* __restrict__ w2a, const float* __restrict__ b2a,   // 0000: [64,32],[64]
    const float* __restrict__ w2b, const float* __restrict__ b2b,   // 0011
    const float* __restrict__ w2c, const float* __restrict__ b2c,   // 1100
    const float* __restrict__ w2d, const float* __restrict__ b2d,   // 1111: [576,32],[576]
    float* __restrict__ outg)         // [E,8,10]
{
    __shared__ float sX0[4][TILE][32];     // trunk act after LN0+lrelu         (8 KB)
    __shared__ float sX2[4][TILE][32];     // after GEMM1 (+b1) then LN1+lrelu  (8 KB)
    __shared__ float sY1111[TILE][576];    // 1111 head output                  (36 KB)
    __shared__ float sYsm[3][TILE][64];    // 0000/0011/1100 head outputs       (12 KB)
    __shared__ float sSrc[TILE][80];       // TDM-gathered h[src_idx]           (5 KB)
    __shared__ float sQ[TILE][8][81];      // Q[a][ri*9+f]                      (40.5 KB)
    __shared__ float sT10[TILE][8];        // sum_ri B1100[ri]*src[i,1+ri]      (0.5 KB)
    __shared__ float sB11[TILE][729];      // TDM-staged basis_11_11 tile       (45.6 KB)

    const int tid = threadIdx.x;
    const int e0  = blockIdx.x * TILE;
    const int w   = tid >> 5;
    const int lane = tid & 31;
    const int n    = lane & 15;   // WMMA N index (or A row index)
    const int hi   = lane >> 4;   // lane half: hi=1 takes K+2,K+3 of each K-step

    // ---- Phase 0 (wave 0 only): launch both TDM transfers up front so the DMA
    //      overlaps the whole MLP/WMMA pipeline. TENSORcnt is per-wave.
    if (w == 0) {
        // (a) Bulk copy: basis_11_11 tile = 16*729 contiguous f32 (46656 B).
        {
            unsigned long long ga = (unsigned long long)(uintptr_t)(B11g + (size_t)e0 * 729);
            unsigned int lb = (unsigned int)(uintptr_t)(&sB11[0][0]);
            u32x4 g0;
            g0[0] = 1u;                                   // count=1, gather off
            g0[1] = lb;                                   // lds_addr
            g0[2] = (unsigned int)ga;                     // global_addr[31:0]
            g0[3] = ((unsigned int)(ga >> 32) & 0x01FFFFFFu) | (2u << 30); // addr[56:32] | type=2
            i32x8 g1;
            g1[0] = 0x00020000;                           // data_size=4B
            g1[1] = (int)(11664u << 16);                  // tensor_dim0[15:0]=11664
            g1[2] = (int)(1u << 16);                      // tensor_dim0 hi=0 | tensor_dim1=1
            g1[3] = (int)(11664u << 16);                  // tensor_dim1 hi=0 | tile_dim0=11664
            g1[4] = 1;                                    // tile_dim1=1, tile_dim2=0
            g1[5] = 11664;                                // tensor_dim0_stride lo
            g1[6] = 0;
            g1[7] = 0;
            i32x4 z = {0, 0, 0, 0};
            tdm_load(g0, g1, z, z);
        }
        // (b) Gather mode: 16 rows of h (80 f32 each) at 16-bit row indices.
        {
            unsigned long long ha = (unsigned long long)(uintptr_t)hg;
            unsigned int ls = (unsigned int)(uintptr_t)(&sSrc[0][0]);
            u32x4 g0;
            g0[0] = 0x80000001u;                          // count=1 | gather_mode, 16-bit idx
            g0[1] = ls;
            g0[2] = (unsigned int)ha;
            g0[3] = ((unsigned int)(ha >> 32) & 0x01FFFFFFu) | (2u << 30);
            i32x8 g1;
            g1[0] = 0x00020000;                           // data_size=4B
            g1[1] = (int)(80u << 16);                     // tensor_dim0=80 (row length)
            g1[2] = (int)(4096u << 16);                   // tensor_dim1=N=4096
            g1[3] = (int)(80u << 16);                     // tile_dim0=80
            g1[4] = 16;                                   // tile_dim1 = #valid indices
            g1[5] = 80;                                   // tensor_dim0_stride = 80
            g1[6] = 0;
            g1[7] = 0;
            int id[16];
            for (int e = 0; e < 16; ++e) id[e] = sidx[e0 + e];
            i32x4 g2, g3;
            for (int j = 0; j < 4; ++j) {
                g2[j] = (id[2 * j] & 0xFFFF) | (id[2 * j + 1] << 16);
                g3[j] = (id[8 + 2 * j] & 0xFFFF) | (id[9 + 2 * j] << 16);
            }
            tdm_load(g0, g1, g2, g3);
        }
    }

    // ---- Phase 1: trunk layer 0  x0 = lrelu(LN(r*w0))  — 64 rows (k,e)
    if (tid < 64) {
        int k = tid >> 4, e = tid & 15;
        float re = rg[e0 + e];
        float xv[32];
        float mu = 0.f;
        for (int c = 0; c < 32; ++c) { xv[c] = re * w0g[k * 32 + c]; mu += xv[c]; }
        mu *= (1.f / 32.f);
        float m2 = 0.f;
        for (int c = 0; c < 32; ++c) { float d = xv[c] - mu; m2 += d * d; }
        float inv = rsqrtf(m2 * (1.f / 32.f) + LN_EPS);
        for (int c = 0; c < 32; ++c) {
            float xn = (xv[c] - mu) * inv * g0g[k * 32 + c] + be0g[k * 32 + c];
            sX0[k][e][c] = lrelu(xn);
        }
    }
    __syncthreads();

    // ---- Phase 2: trunk GEMM  x1 = x0 @ w1[k] + b1   (WMMA f32 16x16x4)
    {
        int k = w >> 1, ct = w & 1;
        int col = ct * 16 + n;
        float bias = b1g[k * 32 + col];
        v8f c;
        for (int v = 0; v < 8; ++v) c[v] = bias;
        for (int kk = 0; kk < 32; kk += 4) {
            v2f a, b;
            a[0] = sX0[k][n][kk + 2 * hi + 0];
            a[1] = sX0[k][n][kk + 2 * hi + 1];
            b[0] = w1g[((k * 32) + col) * 32 + kk + 2 * hi + 0];
            b[1] = w1g[((k * 32) + col) * 32 + kk + 2 * hi + 1];
            c = __builtin_amdgcn_wmma_f32_16x16x4_f32(false, a, false, b, (short)0, c, false, false);
        }
        for (int v = 0; v < 8; ++v) sX2[k][v + 8 * hi][col] = c[v];
    }
    __syncthreads();

    // ---- Phase 2.5: LN1 + lrelu in place on sX2 (64 rows)
    if (tid < 64) {
        int k = tid >> 4, e = tid & 15;
        float mu = 0.f;
        for (int c = 0; c < 32; ++c) mu += sX2[k][e][c];
        mu *= (1.f / 32.f);
        float m2 = 0.f;
        for (int c = 0; c < 32; ++c) { float d = sX2[k][e][c] - mu; m2 += d * d; }
        float inv = rsqrtf(m2 * (1.f / 32.f) + LN_EPS);
        for (int c = 0; c < 32; ++c) {
            float xn = (sX2[k][e][c] - mu) * inv * g1g[k * 32 + c] + be1g[k * 32 + c];
            sX2[k][e][c] = lrelu(xn);
        }
    }
    __syncthreads();

    // ---- Phase 3: head GEMMs (WMMA). 48 col-tiles of 16: T<36 -> 1111 head.
    for (int T = w; T < 48; T += 8) {
        int k, cb, ld;
        const float *wp, *bp;
        float* dst;
        if (T < 36) { k = 3; cb = T * 16; wp = w2d; bp = b2d; dst = &sY1111[0][0]; ld = 576; }
        else {
            int s = T - 36; k = s >> 2; cb = (s & 3) * 16;
            wp  = (k == 0) ? w2a : (k == 1) ? w2b : w2c;
            bp  = (k == 0) ? b2a : (k == 1) ? b2b : b2c;
            dst = &sYsm[k][0][0]; ld = 64;
        }
        int col = cb + n;
        float bias = bp[col];
        v8f c;
        for (int v = 0; v < 8; ++v) c[v] = bias;
        for (int kk = 0; kk < 32; kk += 4) {
            v2f a, b;
            a[0] = sX2[k][n][kk + 2 * hi + 0];
            a[1] = sX2[k][n][kk + 2 * hi + 1];
            b[0] = wp[col * 32 + kk + 2 * hi + 0];
            b[1] = wp[col * 32 + kk + 2 * hi + 1];
            c = __builtin_amdgcn_wmma_f32_16x16x4_f32(false, a, false, b, (short)0, c, false, false);
        }
        for (int v = 0; v < 8; ++v) dst[(size_t)(v + 8 * hi) * ld + col] = c[v];
    }

    // TDM results (sSrc, sB11) are needed from Phase 5 on: issuing wave drains
    // TENSORcnt, then the workgroup barrier publishes the LDS writes to all waves.
    if (w == 0) __builtin_amdgcn_s_wait_tensorcnt(0);
    __syncthreads();

    // ---- Phase 5: Q[e][a][ri*9+f] = sum_i y1111[e][a*72+i*9+f] * src[e][i*10+1+ri]
    //      plus t10[e][i] = sum_ri B1100[e][ri] * src[e][i*10+1+ri]
    {
        int e = tid >> 4, g = tid & 15;
        for (int j = g; j < 648; j += 16) {
            int a = j / 81, mrem = j % 81;
            int ri = mrem / 9, f = mrem % 9;
            float acc = 0.f;
            for (int i = 0; i < 8; ++i)
                acc += sY1111[e][a * 72 + i * 9 + f] * sSrc[e][i * 10 + 1 + ri];
            sQ[e][a][mrem] = acc;
        }
        if (tid < 128) {
            int ee = tid >> 3, i = tid & 7;
            float t = 0.f;
            for (int ri = 0; ri < 9; ++ri)
                t += B10g[(size_t)(e0 + ee) * 9 + ri] * sSrc[ee][i * 10 + 1 + ri];
            sT10[ee][i] = t;
        }
    }
    __syncthreads();

    // ---- Phase 6: outputs. thread (e = tid/16, g = tid%16) handles 5 of the 80 outs.
    {
        int e = tid >> 4, g = tid & 15;
        int ge = e0 + e;
        float B00 = B00g[ge];
        for (int o = g; o < 80; o += 16) {
            int a = o / 10, b = o % 10;
            float val;
            if (b == 0) {
                float s1 = 0.f, s2 = 0.f;
                for (int i = 0; i < 8; ++i) {
                    s1 += sYsm[0][e][a * 8 + i] * sSrc[e][i * 10];
                    s2 += sYsm[2][e][a * 8 + i] * sT10[e][i];
                }
                val = B00 * s1 + s2;
            } else {
                int ro = b - 1;
                float s1 = 0.f;
                for (int i = 0; i < 8; ++i) s1 += sYsm[1][e][a * 8 + i] * sSrc[e][i * 10];
                float s2 = 0.f;
                const float* bp = &sB11[e][ro * 81];
                const float* qp = &sQ[e][a][0];
                for (int m = 0; m < 81; ++m) s2 += qp[m] * bp[m];
                val = B01g[(size_t)ge * 9 + ro] * s1 + s2;
            }
            outg[(size_t)ge * 80 + o] = val;
        }
    }
}

extern "C" void kernel_launch(void* const* d_in, const int* in_sizes, int n_in,
                              void* d_out, int out_size, void* d_ws, size_t ws_size,
                              hipStream_t stream) {
    const float* rg    = (const float*)d_in[0];
    const float* B00g  = (const float*)d_in[1];
    const float* B01g  = (const float*)d_in[2];
    const float* B10g  = (const float*)d_in[3];
    const float* B11g  = (const float*)d_in[4];
    const float* hg    = (const float*)d_in[5];
    const int*   sidx  = (const int*)d_in[6];
    const float* w0g   = (const float*)d_in[7];
    const float* g0g   = (const float*)d_in[8];
    const float* be0g  = (const float*)d_in[9];
    const float* w1g   = (const float*)d_in[10];
    const float* b1g   = (const float*)d_in[11];
    const float* g1g   = (const float*)d_in[12];
    const float* be1g  = (const float*)d_in[13];
    const float* w2a   = (const float*)d_in[14];
    const float* b2a   = (const float*)d_in[15];
    const float* w2b   = (const float*)d_in[16];
    const float* b2b   = (const float*)d_in[17];
    const float* w2c   = (const float*)d_in[18];
    const float* b2c   = (const float*)d_in[19];
    const float* w2d   = (const float*)d_in[20];
    const float* b2d   = (const float*)d_in[21];
    float* outg = (float*)d_out;

    dim3 grid(E_TOT / TILE);   // 2048 workgroups, 16 edges each
    dim3 block(256);           // 8 wave32 waves
    equiconv_fused<<<grid, block, 0, stream>>>(
        rg, B00g, B01g, B10g, B11g, hg, sidx,
        w0g, g0g, be0g, w1g, b1g, g1g, be1g,
        w2a, b2a, w2b, b2b, w2c, b2c, w2d, b2d, outg);
}